// EGSAGEStack_65738769433056
// MI455X (gfx1250) — compile-verified
//
#include <hip/hip_runtime.h>

// EGSAGE 3-layer GNN stack for MI455X (gfx1250), fp32 via V_WMMA_F32_16X16X4_F32.
// GEMMs: M-tile = 16 rows, N = 64 (4 waves x 16), K in {64,128,192}.
// B (weight) fragments are register-resident for the whole kernel (K/4 v2f per
// lane); A tiles are double-buffered in LDS and filled with
// GLOBAL_LOAD_ASYNC_TO_LDS_B128 (ASYNCcnt), overlapping HBM/L2 streaming of
// edge_attr and the L2-resident x[src]/x[tgt] gathers with the WMMA chain.
// wave32 fragment mapping per CDNA5 ISA 7.12.2:
//   A(m,k):  lane = (k>=2)*16 + m, vgpr = k&1
//   B(k,n):  lane = (k>=2)*16 + n, vgpr = k&1
//   C(m,n):  vgpr r -> m = r + 8*(lane>>4), n = (lane&15)

typedef __attribute__((ext_vector_type(2))) float v2f;
typedef __attribute__((ext_vector_type(8))) float v8f;

#define DD 64

// ---- CDNA5 async copy: global -> LDS, 16B per lane, tracked by ASYNCcnt ----
__device__ __forceinline__ void async_ld_b128(void* lds_dst, const void* gaddr)
{
    // Low 32 bits of a generic LDS pointer are the LDS byte address
    // (flat->LDS aperture mapping truncates to addr[31:0]).
    asm volatile("global_load_async_to_lds_b128 %0, %1, off"
                 :: "v"((unsigned)(uintptr_t)lds_dst), "v"(gaddr)
                 : "memory");
}

template<int N>
__device__ __forceinline__ void wait_async()
{
#if __has_builtin(__builtin_amdgcn_s_wait_asynccnt)
    __builtin_amdgcn_s_wait_asynccnt(N);
#else
    asm volatile("s_wait_asynccnt %0" :: "i"(N) : "memory");
#endif
}

// ---- register-resident B fragments + WMMA K-loop ---------------------------
template<int K>
__device__ __forceinline__ void load_bfrag(const float* __restrict__ W,
                                           int n, int kh, v2f* breg)
{
#pragma unroll
    for (int kk = 0; kk < K; kk += 4) {
        const float* w = W + (size_t)(kk + kh) * DD + n;
        v2f b; b.x = w[0]; b.y = w[DD];
        breg[kk >> 2] = b;
    }
}

template<int K>
__device__ __forceinline__ v8f wmma_rowtile_reg(const float* __restrict__ Arow,
                                                const v2f* __restrict__ breg,
                                                int kh)
{
    v8f acc = {};
#pragma unroll
    for (int kk = 0; kk < K; kk += 4) {
        v2f a; a.x = Arow[kk + kh]; a.y = Arow[kk + kh + 1];
        acc = __builtin_amdgcn_wmma_f32_16x16x4_f32(
            /*neg_a=*/false, a, /*neg_b=*/false, breg[kk >> 2],
            /*c_mod=*/(short)0, acc, /*reuse_a=*/false, /*reuse_b=*/false);
    }
    return acc;
}

// ---------------------------------------------------------------- degree ----
__global__ void deg_count_kernel(const int* __restrict__ tgt,
                                 float* __restrict__ deg, int n_edges)
{
    int i = blockIdx.x * blockDim.x + threadIdx.x;
    if (i < n_edges) atomicAdd(deg + tgt[i], 1.0f);
}

__global__ void deg_inv_kernel(float* __restrict__ deg, int n_nodes)
{
    int i = blockIdx.x * blockDim.x + threadIdx.x;
    if (i < n_nodes) deg[i] = 1.0f / fmaxf(deg[i], 1.0f);
}

// ------------------------------------------------ message GEMM + scatter ----
// m = relu([x[src] | edge_attr] @ W (128x64) + b) ; atomicAdd into agg[tgt]
__global__ void __launch_bounds__(128, 1)
egsage_msg_scatter(const float* __restrict__ x, const float* __restrict__ ea,
                   const int* __restrict__ src, const int* __restrict__ tgt,
                   const float* __restrict__ W, const float* __restrict__ bias,
                   float* __restrict__ agg, int n_edges)
{
    __shared__ float sA[2][16 * 128];   // double-buffered A tiles (2 x 8 KB)
    __shared__ int   sT[2][16];

    const int tid  = threadIdx.x;
    const int lane = tid & 31;
    const int wv   = tid >> 5;
    const int m    = lane & 15;
    const int kh   = (lane >> 4) << 1;
    const int n    = (wv << 4) + (lane & 15);
    const int r    = tid >> 3;           // staging row 0..15
    const int cq   = tid & 7;            // staging float4 group 0..7

    v2f breg[32];                        // 64 VGPRs of weights, kernel-resident
    load_bfrag<128>(W, n, kh, breg);
    const float bn = bias[n];

    const int ntile = (n_edges + 15) >> 4;

    auto issue = [&](int tile, int buf) {
        int e = tile * 16 + r; if (e >= n_edges) e = n_edges - 1;
        const int s = src[e];
        const float4* xr = (const float4*)(x  + (size_t)s * DD);
        const float4* er = (const float4*)(ea + (size_t)e * DD);
        float4* a0 = (float4*)(sA[buf] + r * 128);
        float4* a1 = a0 + 16;            // +64 floats
        async_ld_b128(a0 + cq,     xr + cq);      // 4 async ops/wave/tile
        async_ld_b128(a0 + cq + 8, xr + cq + 8);
        async_ld_b128(a1 + cq,     er + cq);
        async_ld_b128(a1 + cq + 8, er + cq + 8);
        if (tid < 16) {
            int et = tile * 16 + tid;
            sT[buf][tid] = (et < n_edges) ? tgt[et] : -1;
        }
    };

    int tile = blockIdx.x;
    if (tile < ntile) issue(tile, 0);
    int buf = 0;
    for (; tile < ntile; tile += gridDim.x, buf ^= 1) {
        const int nt = tile + gridDim.x;
        if (nt < ntile) { issue(nt, buf ^ 1); wait_async<4>(); }
        else            { wait_async<0>(); }
        __syncthreads();                 // buf tile fully in LDS for all waves

        v8f acc = wmma_rowtile_reg<128>(sA[buf] + m * 128, breg, kh);

#pragma unroll
        for (int rr = 0; rr < 8; ++rr) {
            const int row = rr + ((lane >> 4) << 3);
            const int t = sT[buf][row];
            if (t >= 0) {
                float v = acc[rr] + bn;
                v = v > 0.0f ? v : 0.0f;
                atomicAdd(agg + (size_t)t * DD + n, v);   // L2-resident target
            }
        }
        __syncthreads();                 // done reading buf before refill
    }
}

// ------------------------------------------- node update GEMM + L2-norm ----
// h = relu([agg/deg | x] @ W (128x64) + b) ; x' = h / max(||h||,1e-12)
__global__ void __launch_bounds__(128, 1)
egsage_node_update(const float* __restrict__ agg, const float* __restrict__ dinv,
                   const float* xin,                 // may alias xout (in place)
                   const float* __restrict__ W, const float* __restrict__ bias,
                   float* xout, int n_nodes)
{
    __shared__ float sA[16 * 128];
    __shared__ float sO[16 * DD];
    __shared__ float sInv[16];

    const int tid  = threadIdx.x;
    const int lane = tid & 31;
    const int wv   = tid >> 5;
    const int m    = lane & 15;
    const int kh   = (lane >> 4) << 1;
    const int n    = (wv << 4) + (lane & 15);
    const int r    = tid >> 3;
    const int cq   = tid & 7;

    v2f breg[32];
    load_bfrag<128>(W, n, kh, breg);
    const float bn = bias[n];

    const int ntile = (n_nodes + 15) >> 4;
    for (int tile = blockIdx.x; tile < ntile; tile += gridDim.x) {
        const int n0 = tile << 4;
        {
            int node = n0 + r; if (node >= n_nodes) node = n_nodes - 1;
            const float di = dinv[node];
            const float4* gr = (const float4*)(agg + (size_t)node * DD);
            const float4* xr = (const float4*)(xin + (size_t)node * DD);
            float4* a0 = (float4*)(sA + r * 128);
            float4* a1 = a0 + 16;
            float4 g0 = gr[cq], g1 = gr[cq + 8];
            g0.x *= di; g0.y *= di; g0.z *= di; g0.w *= di;
            g1.x *= di; g1.y *= di; g1.z *= di; g1.w *= di;
            a0[cq] = g0; a0[cq + 8] = g1;
            a1[cq] = xr[cq]; a1[cq + 8] = xr[cq + 8];
        }
        __syncthreads();

        v8f acc = wmma_rowtile_reg<128>(sA + m * 128, breg, kh);

#pragma unroll
        for (int rr = 0; rr < 8; ++rr) {
            const int row = rr + ((lane >> 4) << 3);
            float v = acc[rr] + bn;
            sO[row * DD + n] = v > 0.0f ? v : 0.0f;
        }
        __syncthreads();

        if (tid < 16) {
            float ss = 0.0f;
            const float* h = sO + tid * DD;
#pragma unroll 8
            for (int j = 0; j < DD; ++j) ss += h[j] * h[j];
            sInv[tid] = 1.0f / fmaxf(sqrtf(ss), 1e-12f);
        }
        __syncthreads();

        {
            const int base = tid * 8;    // 16*64/128 = 8 floats/thread
            const int row = base >> 6;
            if (n0 + row < n_nodes) {
                const float s = sInv[row];
                float4 o0 = ((const float4*)(sO + base))[0];
                float4 o1 = ((const float4*)(sO + base))[1];
                o0.x *= s; o0.y *= s; o0.z *= s; o0.w *= s;
                o1.x *= s; o1.y *= s; o1.z *= s; o1.w *= s;
                float4* dst = (float4*)(xout + (size_t)n0 * DD + base);
                dst[0] = o0; dst[1] = o1;
            }
        }
        __syncthreads();
    }
}

// ---------------------------------------------------- edge update GEMM -----
// e' = relu([x[src] | x[tgt] | e] @ W (192x64) + b), written in place
__global__ void __launch_bounds__(128, 1)
egsage_edge_update(const float* __restrict__ x,
                   const int* __restrict__ src, const int* __restrict__ tgt,
                   const float* ein,               // may alias eout (in place)
                   const float* __restrict__ W, const float* __restrict__ bias,
                   float* eout, int n_edges)
{
    __shared__ float sA[2][16 * 192];   // double-buffered (2 x 12 KB)

    const int tid  = threadIdx.x;
    const int lane = tid & 31;
    const int wv   = tid >> 5;
    const int m    = lane & 15;
    const int kh   = (lane >> 4) << 1;
    const int n    = (wv << 4) + (lane & 15);
    const int r    = tid >> 3;
    const int cq   = tid & 7;

    v2f breg[48];                        // 96 VGPRs of weights, kernel-resident
    load_bfrag<192>(W, n, kh, breg);
    const float bn = bias[n];

    const int ntile = (n_edges + 15) >> 4;

    auto issue = [&](int tile, int buf) {
        int e = tile * 16 + r; if (e >= n_edges) e = n_edges - 1;
        const int s = src[e], t = tgt[e];
        const float4* xs = (const float4*)(x   + (size_t)s * DD);
        const float4* xt = (const float4*)(x   + (size_t)t * DD);
        const float4* er = (const float4*)(ein + (size_t)e * DD);
        float4* a0 = (float4*)(sA[buf] + r * 192);
        float4* a1 = a0 + 16;
        float4* a2 = a0 + 32;
        async_ld_b128(a0 + cq,     xs + cq);      // 6 async ops/wave/tile
        async_ld_b128(a0 + cq + 8, xs + cq + 8);
        async_ld_b128(a1 + cq,     xt + cq);
        async_ld_b128(a1 + cq + 8, xt + cq + 8);
        async_ld_b128(a2 + cq,     er + cq);
        async_ld_b128(a2 + cq + 8, er + cq + 8);
    };

    int tile = blockIdx.x;
    if (tile < ntile) issue(tile, 0);
    int buf = 0;
    for (; tile < ntile; tile += gridDim.x, buf ^= 1) {
        const int nt = tile + gridDim.x;
        if (nt < ntile) { issue(nt, buf ^ 1); wait_async<6>(); }
        else            { wait_async<0>(); }
        __syncthreads();

        v8f acc = wmma_rowtile_reg<192>(sA[buf] + m * 192, breg, kh);

        const int e0 = tile << 4;
#pragma unroll
        for (int rr = 0; rr < 8; ++rr) {
            const int row = rr + ((lane >> 4) << 3);
            const int e = e0 + row;
            if (e < n_edges) {
                float v = acc[rr] + bn;
                eout[(size_t)e * DD + n] = v > 0.0f ? v : 0.0f;
            }
        }
        __syncthreads();
    }
}

// -------------------------------------------------------- post-MLP ---------
// out = relu(x @ W1 + b1) @ W2 + b2, both 64x64, fused
__global__ void __launch_bounds__(128, 1)
egsage_post(const float* __restrict__ x,
            const float* __restrict__ W1, const float* __restrict__ b1,
            const float* __restrict__ W2, const float* __restrict__ b2,
            float* __restrict__ out, int n_nodes)
{
    __shared__ float sA[16 * DD];
    __shared__ float sH[16 * DD];

    const int tid  = threadIdx.x;
    const int lane = tid & 31;
    const int wv   = tid >> 5;
    const int m    = lane & 15;
    const int kh   = (lane >> 4) << 1;
    const int n    = (wv << 4) + (lane & 15);
    const int r    = tid >> 3;
    const int cq   = tid & 7;

    v2f breg1[16], breg2[16];
    load_bfrag<64>(W1, n, kh, breg1);
    load_bfrag<64>(W2, n, kh, breg2);
    const float b1n = b1[n], b2n = b2[n];

    const int ntile = (n_nodes + 15) >> 4;
    for (int tile = blockIdx.x; tile < ntile; tile += gridDim.x) {
        const int n0 = tile << 4;
        {
            int node = n0 + r; if (node >= n_nodes) node = n_nodes - 1;
            const float4* xr = (const float4*)(x + (size_t)node * DD);
            float4* a = (float4*)(sA + r * DD);
            a[cq] = xr[cq]; a[cq + 8] = xr[cq + 8];
        }
        __syncthreads();

        v8f h = wmma_rowtile_reg<64>(sA + m * DD, breg1, kh);
#pragma unroll
        for (int rr = 0; rr < 8; ++rr) {
            const int row = rr + ((lane >> 4) << 3);
            float v = h[rr] + b1n;
            sH[row * DD + n] = v > 0.0f ? v : 0.0f;
        }
        __syncthreads();

        v8f o = wmma_rowtile_reg<64>(sH + m * DD, breg2, kh);
#pragma unroll
        for (int rr = 0; rr < 8; ++rr) {
            const int row = rr + ((lane >> 4) << 3);
            if (n0 + row < n_nodes)
                out[(size_t)(n0 + row) * DD + n] = o[rr] + b2n;
        }
        __syncthreads();
    }
}

// ---------------------------------------------------------------------------
extern "C" void kernel_launch(void* const* d_in, const int* in_sizes, int n_in,
                              void* d_out, int out_size, void* d_ws, size_t ws_size,
                              hipStream_t stream)
{
    const float* x_in  = (const float*)d_in[0];
    const float* ea_in = (const float*)d_in[1];
    const int*   eidx  = (const int*)  d_in[2];
    const float* msgW  = (const float*)d_in[3];
    const float* msgB  = (const float*)d_in[4];
    const float* aggW  = (const float*)d_in[5];
    const float* aggB  = (const float*)d_in[6];
    const float* eupW  = (const float*)d_in[7];
    const float* eupB  = (const float*)d_in[8];
    const float* pW1   = (const float*)d_in[9];
    const float* pB1   = (const float*)d_in[10];
    const float* pW2   = (const float*)d_in[11];
    const float* pB2   = (const float*)d_in[12];
    float*       out   = (float*)d_out;

    const int n_nodes = in_sizes[0] / DD;
    const int n_edges = in_sizes[2] / 2;
    const int* src = eidx;
    const int* tgt = eidx + n_edges;

    // workspace carve-up (all offsets 256B-aligned)
    float* ws  = (float*)d_ws;
    size_t o   = 0;
    float* deg = ws + o; o += (size_t)((n_nodes + 255) & ~255);
    float* agg = ws + o; o += (size_t)n_nodes * DD;
    float* X   = ws + o; o += (size_t)n_nodes * DD;
    float* E   = ws + o; o += (size_t)n_edges * DD;
    (void)ws_size; (void)n_in; (void)out_size;

    const int etiles = (n_edges + 15) >> 4;
    const int ntiles = (n_nodes + 15) >> 4;
    const int eblocks = etiles < 8192 ? etiles : 8192;
    const int nblocks = ntiles < 8192 ? ntiles : 8192;

    // degree -> inverse degree (recomputed each call: deterministic, cheap)
    hipMemsetAsync(deg, 0, sizeof(float) * (size_t)n_nodes, stream);
    deg_count_kernel<<<(n_edges + 255) / 256, 256, 0, stream>>>(tgt, deg, n_edges);
    deg_inv_kernel<<<(n_nodes + 255) / 256, 256, 0, stream>>>(deg, n_nodes);

    for (int l = 0; l < 3; ++l) {
        const float* xl = (l == 0) ? x_in  : X;   // node features this layer
        const float* el = (l == 0) ? ea_in : E;   // edge features this layer

        hipMemsetAsync(agg, 0, sizeof(float) * (size_t)n_nodes * DD, stream);

        egsage_msg_scatter<<<eblocks, 128, 0, stream>>>(
            xl, el, src, tgt, msgW + (size_t)l * 128 * DD, msgB + (size_t)l * DD,
            agg, n_edges);

        egsage_node_update<<<nblocks, 128, 0, stream>>>(
            agg, deg, xl, aggW + (size_t)l * 128 * DD, aggB + (size_t)l * DD,
            X, n_nodes);

        // in-place for l>0 is safe: each edge row's async reads complete
        // before the barrier that precedes its stores
        egsage_edge_update<<<eblocks, 128, 0, stream>>>(
            X, src, tgt, el, eupW + (size_t)l * 192 * DD, eupB + (size_t)l * DD,
            E, n_edges);
    }

    egsage_post<<<nblocks, 128, 0, stream>>>(X, pW1, pB1, pW2, pB2, out, n_nodes);
}